// MTGNN_31215822307351
// MI455X (gfx1250) — compile-verified
//
#include <hip/hip_runtime.h>
#include <hip/hip_bf16.h>
#include <math.h>

// ---------------------------------------------------------------------------
// MTGNN forward for MI455X (gfx1250). fp32 end-to-end.
// Heavy op = graph propagation GEMM (G[2048x2048] x F[2048xM]) done with
// V_WMMA_F32_16X16X4_F32 (wave32 matrix pipe, fp32-accurate), 1x2 register
// blocking per wave; both A and B tiles stored so each lane's fragment is a
// contiguous dword pair in LDS (no VGPR repacking movs).
// ---------------------------------------------------------------------------

typedef __attribute__((ext_vector_type(2))) float v2f;
typedef __attribute__((ext_vector_type(8))) float v8f;

static constexpr int B_ = 32, N_ = 2048, TIN_ = 12, IN_DIM_ = 2;
static constexpr int RES_C_ = 32, CONV_C_ = 32, SKIP_C_ = 64, END_C_ = 128, OUT_DIM_ = 12;
static constexpr int NODE_DIM_ = 40, K_SUB_ = 20, RF_ = 19;
static constexpr float ALPHA_ = 0.05f, TANH_A_ = 3.0f, EPS_ = 1e-5f;

#define CDIV(a, b) (((a) + (b) - 1) / (b))

// ------------------------------ graph build --------------------------------

__global__ void k_nodevec(const float* __restrict__ emb1, const float* __restrict__ emb2,
                          const int* __restrict__ idx,
                          const float* __restrict__ w1, const float* __restrict__ b1,
                          const float* __restrict__ w2, const float* __restrict__ b2,
                          float* __restrict__ nv1, float* __restrict__ nv2) {
    int t = blockIdx.x * blockDim.x + threadIdx.x;
    if (t >= N_ * NODE_DIM_) return;
    int n = t / NODE_DIM_, d = t % NODE_DIM_;
    int e = idx[n];
    float a1 = b1[d], a2 = b2[d];
    for (int k = 0; k < NODE_DIM_; ++k) {
        a1 += emb1[e * NODE_DIM_ + k] * w1[d * NODE_DIM_ + k];
        a2 += emb2[e * NODE_DIM_ + k] * w2[d * NODE_DIM_ + k];
    }
    nv1[n * NODE_DIM_ + d] = tanhf(TANH_A_ * a1);
    nv2[n * NODE_DIM_ + d] = tanhf(TANH_A_ * a2);
}

// One block per row i: adj row, top-20 selection (20 deterministic argmax
// rounds), masked write, and row sum (+1 for the identity) for normalization.
__global__ void k_adj_row(const float* __restrict__ nv1, const float* __restrict__ nv2,
                          float* __restrict__ adjm, float* __restrict__ rowsum1) {
    __shared__ float rowv[N_];
    __shared__ float v1[NODE_DIM_], v2[NODE_DIM_];
    __shared__ float smax[256];
    __shared__ int   sidx[256];
    __shared__ float ssum;
    int i = blockIdx.x, tid = threadIdx.x;
    if (tid < NODE_DIM_) { v1[tid] = nv1[i * NODE_DIM_ + tid]; v2[tid] = nv2[i * NODE_DIM_ + tid]; }
    if (tid == 0) ssum = 0.f;
    __syncthreads();
    for (int j = tid; j < N_; j += 256) {
        float d1 = 0.f, d2 = 0.f;
        for (int k = 0; k < NODE_DIM_; ++k) {
            d1 += v1[k] * nv2[j * NODE_DIM_ + k];
            d2 += v2[k] * nv1[j * NODE_DIM_ + k];
        }
        float a = tanhf(TANH_A_ * (d1 - d2));
        rowv[j] = a > 0.f ? a : 0.f;
    }
    __syncthreads();
    for (int r = 0; r < K_SUB_; ++r) {               // selected entries encoded as -(v)-10
        float m = -1.f; int mi = 0;
        for (int j = tid; j < N_; j += 256) { float v = rowv[j]; if (v > m) { m = v; mi = j; } }
        smax[tid] = m; sidx[tid] = mi;
        __syncthreads();
        for (int s = 128; s > 0; s >>= 1) {
            if (tid < s && smax[tid + s] > smax[tid]) { smax[tid] = smax[tid + s]; sidx[tid] = sidx[tid + s]; }
            __syncthreads();
        }
        if (tid == 0) { ssum += smax[0]; rowv[sidx[0]] = -smax[0] - 10.f; }
        __syncthreads();
    }
    for (int j = tid; j < N_; j += 256) {
        float v = rowv[j];
        adjm[(long)i * N_ + j] = (v < -5.f) ? (-(v + 10.f)) : 0.f;
    }
    if (tid == 0) rowsum1[i] = ssum + 1.f;
}

__global__ void k_colsum(const float* __restrict__ adjm, float* __restrict__ colsum) {
    int v = blockIdx.x * blockDim.x + threadIdx.x;
    if (v >= N_) return;
    float s = 0.f;
    for (int w = 0; w < N_; ++w) s += adjm[(long)w * N_ + v];
    colsum[v] = s + 1.f;
}

// G1[w][v] = rownorm(adj+I)[v][w],  G2[w][v] = rownorm(adj^T+I)[v][w]
__global__ void k_buildG(const float* __restrict__ adjm, const float* __restrict__ rowsum1,
                         const float* __restrict__ colsum,
                         float* __restrict__ G1, float* __restrict__ G2) {
    long t = (long)blockIdx.x * blockDim.x + threadIdx.x;
    if (t >= (long)N_ * N_) return;
    int w = (int)(t / N_), v = (int)(t % N_);
    float d = (v == w) ? 1.f : 0.f;
    G1[t] = (adjm[(long)v * N_ + w] + d) / rowsum1[v];
    G2[t] = (adjm[(long)w * N_ + v] + d) / colsum[v];
}

// ------------------------------ stem ---------------------------------------

__global__ void k_start(const float* __restrict__ x, const float* __restrict__ sw,
                        const float* __restrict__ sb, float* __restrict__ hA) {
    long t = (long)blockIdx.x * blockDim.x + threadIdx.x;
    if (t >= (long)B_ * RES_C_ * N_ * RF_) return;
    int tt = (int)(t % RF_); long r = t / RF_;
    int n = (int)(r % N_); r /= N_;
    int c = (int)(r % RES_C_); int b = (int)(r / RES_C_);
    float v = sb[c];
    if (tt >= RF_ - TIN_) {
        int tx = tt - (RF_ - TIN_);
        v += sw[c * 2 + 0] * x[(((long)b * IN_DIM_ + 0) * N_ + n) * TIN_ + tx]
           + sw[c * 2 + 1] * x[(((long)b * IN_DIM_ + 1) * N_ + n) * TIN_ + tx];
    }
    hA[t] = v;
}

__global__ void k_skip0(const float* __restrict__ x, const float* __restrict__ w,
                        const float* __restrict__ bias, float* __restrict__ skip) {
    int t = blockIdx.x * blockDim.x + threadIdx.x;
    if (t >= B_ * SKIP_C_ * N_) return;
    int n = t % N_; int r = t / N_;
    int co = r % SKIP_C_; int b = r / SKIP_C_;
    float acc = bias[co];
    for (int cin = 0; cin < IN_DIM_; ++cin)
        for (int tx = 0; tx < TIN_; ++tx)
            acc += w[(co * IN_DIM_ + cin) * RF_ + (tx + RF_ - TIN_)]
                 * x[(((long)b * IN_DIM_ + cin) * N_ + n) * TIN_ + tx];
    skip[t] = acc;
}

// --------------------------- inception + gate ------------------------------

struct IncepArgs { const float* fw[4]; const float* fb[4]; const float* gw[4]; const float* gb[4]; };

__global__ void k_incep(const float* __restrict__ hin, float* __restrict__ hout,
                        IncepArgs A, int Tin, int Tout) {
    long t = (long)blockIdx.x * blockDim.x + threadIdx.x;
    if (t >= (long)B_ * CONV_C_ * N_ * Tout) return;
    int to = (int)(t % Tout); long r = t / Tout;
    int n = (int)(r % N_); r /= N_;
    int c = (int)(r % CONV_C_); int b = (int)(r / CONV_C_);
    const int KS[4] = {2, 3, 6, 7};
    int br = c >> 3, k = KS[br], cb = c & 7;
    const float* fw = A.fw[br]; const float* gw = A.gw[br];
    float f = A.fb[br][cb], g = A.gb[br][cb];
    int t0 = to + 7 - k;                       // alignment to the shortest branch
    for (int cin = 0; cin < RES_C_; ++cin) {
        const float* hp  = hin + (((long)b * RES_C_ + cin) * N_ + n) * Tin + t0;
        const float* fwp = fw + (cb * RES_C_ + cin) * k;
        const float* gwp = gw + (cb * RES_C_ + cin) * k;
        for (int tap = 0; tap < k; ++tap) { float hv = hp[tap]; f += fwp[tap] * hv; g += gwp[tap] * hv; }
    }
    hout[t] = tanhf(f) * (1.f / (1.f + expf(-g)));
}

__global__ void k_skip_accum(const float* __restrict__ h, const float* __restrict__ w,
                             const float* __restrict__ bias, float* __restrict__ skip, int T) {
    int t = blockIdx.x * blockDim.x + threadIdx.x;
    if (t >= B_ * SKIP_C_ * N_) return;
    int n = t % N_; int r = t / N_;
    int co = r % SKIP_C_; int b = r / SKIP_C_;
    float acc = bias[co];
    for (int cin = 0; cin < CONV_C_; ++cin) {
        const float* hp = h + (((long)b * CONV_C_ + cin) * N_ + n) * T;
        const float* wp = w + (co * CONV_C_ + cin) * T;
        for (int tt = 0; tt < T; ++tt) acc += wp[tt] * hp[tt];
    }
    skip[t] += acc;
}

// pack h[B,C,N,T] -> F[v][(b*T+t)*32 + c]  (node-major for the GEMM)
__global__ void k_pack(const float* __restrict__ h, float* __restrict__ F, int T) {
    long t = (long)blockIdx.x * blockDim.x + threadIdx.x;
    if (t >= (long)B_ * CONV_C_ * N_ * T) return;
    int c = (int)(t % CONV_C_); long r = t / CONV_C_;
    int tt = (int)(r % T); r /= T;
    int b = (int)(r % B_); int v = (int)(r / B_);
    F[t] = h[(((long)b * CONV_C_ + c) * N_ + v) * T + tt];
}

// --------------------- propagation GEMM (fp32 WMMA) ------------------------
// Out[w,m] = ALPHA*X[w,m] + (1-ALPHA) * sum_v G[w,v]*Bm[v,m]
// 256 threads = 8 wave32s; 64x64 output tile/block; each wave owns a 16x32
// sub-tile (two f32 accumulators sharing one A fragment). A tile row-major,
// B tile column-major in LDS so every lane's fragment is a contiguous dword
// pair (single ds load into an aligned VGPR pair, no repacking movs).
// Stride 65 keeps both staging and fragment reads bank-conflict-free.
__global__ __launch_bounds__(256)
void k_gemm_prop(const float* __restrict__ G, const float* __restrict__ Bm,
                 const float* __restrict__ X, float* __restrict__ Out, int M) {
    __shared__ float Gs[64][65];    // Gs[row][k]
    __shared__ float FsT[64][65];   // FsT[col][k]  (transposed B tile)
    int tid = threadIdx.x;
    int wv = tid >> 5, lane = tid & 31;
    int half = lane >> 4, lm = lane & 15;
    int rowofs = (wv & 3) * 16, colofs = (wv >> 2) * 32;
    long wrow0 = (long)blockIdx.x * 64;
    long mcol0 = (long)blockIdx.y * 64;
    v8f acc0 = {}, acc1 = {};
    for (int k0 = 0; k0 < N_; k0 += 64) {
        for (int q = tid; q < 64 * 64; q += 256) {        // stage 64x64 of G
            int r = q >> 6, cidx = q & 63;
            Gs[r][cidx] = G[(wrow0 + r) * N_ + k0 + cidx];
        }
        for (int q = tid; q < 64 * 64; q += 256) {        // stage 64x64 of Bm, transposed
            int r = q >> 6, cidx = q & 63;
            FsT[cidx][r] = Bm[(long)(k0 + r) * M + mcol0 + cidx];
        }
        if (k0 + 64 < N_)                                 // global_prefetch_b8
            __builtin_prefetch(&G[(wrow0 + (tid & 63)) * N_ + k0 + 64], 0, 1);
        __syncthreads();
#pragma unroll
        for (int kk = 0; kk < 16; ++kk) {
            int kb = kk * 4;
            v2f a, b0, b1;
            // A 16x4: lanes0-15 -> K={kb,kb+1}; lanes16-31 -> K={kb+2,kb+3}
            a.x = Gs[rowofs + lm][kb + 2 * half];
            a.y = Gs[rowofs + lm][kb + 2 * half + 1];
            // B 4x16: n = lane&15; K walks down VGPRs within each lane half
            b0.x = FsT[colofs + lm][kb + 2 * half];
            b0.y = FsT[colofs + lm][kb + 2 * half + 1];
            b1.x = FsT[colofs + 16 + lm][kb + 2 * half];
            b1.y = FsT[colofs + 16 + lm][kb + 2 * half + 1];
            acc0 = __builtin_amdgcn_wmma_f32_16x16x4_f32(false, a, false, b0,
                                                         (short)0, acc0, false, false);
            acc1 = __builtin_amdgcn_wmma_f32_16x16x4_f32(false, a, false, b1,
                                                         (short)0, acc1, false, false);
        }
        __syncthreads();
    }
#pragma unroll
    for (int r = 0; r < 8; ++r) {           // C/D: VGPR r -> M=r (lanes0-15) / r+8
        long w = wrow0 + rowofs + r + half * 8;
        long m = mcol0 + colofs + lm;
        Out[w * M + m]      = ALPHA_ * X[w * M + m]      + (1.f - ALPHA_) * acc0[r];
        Out[w * M + m + 16] = ALPHA_ * X[w * M + m + 16] + (1.f - ALPHA_) * acc1[r];
    }
}

// 1x1 conv over concat([F0,H1,H2]) channels; 2nd call accumulates + residual.
__global__ void k_mixconv(const float* __restrict__ F0, const float* __restrict__ H1,
                          const float* __restrict__ H2, const float* __restrict__ gw,
                          const float* __restrict__ gb, float* __restrict__ dst,
                          const float* __restrict__ resid, int T, int Tin, int accum) {
    long t = (long)blockIdx.x * blockDim.x + threadIdx.x;
    if (t >= (long)B_ * RES_C_ * N_ * T) return;
    int tt = (int)(t % T); long r = t / T;
    int n = (int)(r % N_); r /= N_;
    int c = (int)(r % RES_C_); int b = (int)(r / RES_C_);
    long base = (((long)n * B_ + b) * T + tt) * CONV_C_;
    float acc = gb[c];
    const float* w0 = gw + c * 96;
    for (int q = 0; q < 32; ++q)
        acc += w0[q] * F0[base + q] + w0[32 + q] * H1[base + q] + w0[64 + q] * H2[base + q];
    if (accum) {
        float res = resid[(((long)b * RES_C_ + c) * N_ + n) * Tin + (tt + Tin - T)];
        dst[t] += acc + res;
    } else {
        dst[t] = acc;
    }
}

// ------------------------------ layernorm ----------------------------------

__global__ void k_ln_reduce(const float* __restrict__ y, float* __restrict__ part, int T) {
    int b = blockIdx.x >> 8, chunk = blockIdx.x & 255;
    long E = (long)RES_C_ * N_ * T;
    const float* yb = y + (long)b * E;
    float s = 0.f, s2 = 0.f;
    for (long i = (long)chunk * 256 + threadIdx.x; i < E; i += 256 * 256) {
        float v = yb[i]; s += v; s2 += v * v;
    }
    __shared__ float ss[256], ss2[256];
    ss[threadIdx.x] = s; ss2[threadIdx.x] = s2;
    __syncthreads();
    for (int st = 128; st > 0; st >>= 1) {
        if (threadIdx.x < st) { ss[threadIdx.x] += ss[threadIdx.x + st]; ss2[threadIdx.x] += ss2[threadIdx.x + st]; }
        __syncthreads();
    }
    if (threadIdx.x == 0) { part[(b * 256 + chunk) * 2] = ss[0]; part[(b * 256 + chunk) * 2 + 1] = ss2[0]; }
}

__global__ void k_ln_final(const float* __restrict__ part, float* __restrict__ stats, int T) {
    int b = blockIdx.x;
    __shared__ float ss[256], ss2[256];
    ss[threadIdx.x] = part[(b * 256 + threadIdx.x) * 2];
    ss2[threadIdx.x] = part[(b * 256 + threadIdx.x) * 2 + 1];
    __syncthreads();
    for (int st = 128; st > 0; st >>= 1) {
        if (threadIdx.x < st) { ss[threadIdx.x] += ss[threadIdx.x + st]; ss2[threadIdx.x] += ss2[threadIdx.x + st]; }
        __syncthreads();
    }
    if (threadIdx.x == 0) {
        float E = (float)((long)RES_C_ * N_ * T);
        float mu = ss[0] / E;
        stats[b * 2] = mu;
        stats[b * 2 + 1] = ss2[0] / E - mu * mu;
    }
}

__global__ void k_ln_apply(const float* __restrict__ y, const float* __restrict__ stats,
                           const float* __restrict__ nw, const float* __restrict__ nb,
                           const int* __restrict__ idx, float* __restrict__ dst, int T) {
    long t = (long)blockIdx.x * blockDim.x + threadIdx.x;
    if (t >= (long)B_ * RES_C_ * N_ * T) return;
    int tt = (int)(t % T); long r = t / T;
    int n = (int)(r % N_); r /= N_;
    int c = (int)(r % RES_C_); int b = (int)(r / RES_C_);
    float xn = (y[t] - stats[b * 2]) * rsqrtf(stats[b * 2 + 1] + EPS_);
    long a = ((long)c * N_ + idx[n]) * T + tt;
    dst[t] = xn * nw[a] + nb[a];
}

// -------------------------------- head -------------------------------------

__global__ void k_head_skip(const float* __restrict__ h, const float* __restrict__ w,
                            const float* __restrict__ bias, float* __restrict__ skip) {
    int t = blockIdx.x * blockDim.x + threadIdx.x;
    if (t >= B_ * SKIP_C_ * N_) return;
    int n = t % N_; int r = t / N_;
    int co = r % SKIP_C_; int b = r / SKIP_C_;
    float acc = skip[t] + bias[co];
    for (int c = 0; c < RES_C_; ++c) acc += w[co * RES_C_ + c] * h[((long)b * RES_C_ + c) * N_ + n];
    skip[t] = acc > 0.f ? acc : 0.f;
}

__global__ void k_end1(const float* __restrict__ s, const float* __restrict__ w,
                       const float* __restrict__ bias, float* __restrict__ e1) {
    int t = blockIdx.x * blockDim.x + threadIdx.x;
    if (t >= B_ * END_C_ * N_) return;
    int n = t % N_; int r = t / N_;
    int co = r % END_C_; int b = r / END_C_;
    float acc = bias[co];
    for (int c = 0; c < SKIP_C_; ++c) acc += w[co * SKIP_C_ + c] * s[((long)b * SKIP_C_ + c) * N_ + n];
    e1[t] = acc > 0.f ? acc : 0.f;
}

__global__ void k_end2(const float* __restrict__ e1, const float* __restrict__ w,
                       const float* __restrict__ bias, float* __restrict__ out) {
    int t = blockIdx.x * blockDim.x + threadIdx.x;
    if (t >= B_ * OUT_DIM_ * N_) return;
    int n = t % N_; int r = t / N_;
    int o = r % OUT_DIM_; int b = r / OUT_DIM_;
    float acc = bias[o];
    for (int c = 0; c < END_C_; ++c) acc += w[o * END_C_ + c] * e1[((long)b * END_C_ + c) * N_ + n];
    out[t] = acc;
}

// ----------------------------- launcher ------------------------------------

extern "C" void kernel_launch(void* const* d_in, const int* in_sizes, int n_in,
                              void* d_out, int out_size, void* d_ws, size_t ws_size,
                              hipStream_t stream) {
    (void)in_sizes; (void)n_in; (void)out_size; (void)ws_size;
    auto P = [&](int i) { return (const float*)d_in[i]; };
    const float* x   = P(0);
    const int*   idx = (const int*)d_in[1];

    // carve workspace (~810 MB of fp32)
    float* f = (float*)d_ws;
    auto alloc = [&](long nf) { float* p = f; f += nf; return p; };
    float* nv1   = alloc((long)N_ * NODE_DIM_);
    float* nv2   = alloc((long)N_ * NODE_DIM_);
    float* adjm  = alloc((long)N_ * N_);
    float* rs1   = alloc(N_);
    float* cs    = alloc(N_);
    float* G1    = alloc((long)N_ * N_);
    float* G2    = alloc((long)N_ * N_);
    float* hA    = alloc((long)B_ * RES_C_ * N_ * RF_);
    float* hB    = alloc((long)B_ * RES_C_ * N_ * 13);
    float* hC    = alloc((long)B_ * RES_C_ * N_ * 13);
    float* F0    = alloc((long)N_ * B_ * 13 * CONV_C_);
    float* H1    = alloc((long)N_ * B_ * 13 * CONV_C_);
    float* H2    = alloc((long)N_ * B_ * 13 * CONV_C_);
    float* skip  = alloc((long)B_ * SKIP_C_ * N_);
    float* e1    = alloc((long)B_ * END_C_ * N_);
    float* part  = alloc((long)B_ * 256 * 2);
    float* stats = alloc((long)B_ * 2);

    const dim3 TB(256);
    // graph
    k_nodevec<<<CDIV(N_ * NODE_DIM_, 256), TB, 0, stream>>>(P(2), P(3), idx, P(4), P(5), P(6), P(7), nv1, nv2);
    k_adj_row<<<N_, TB, 0, stream>>>(nv1, nv2, adjm, rs1);
    k_colsum<<<CDIV(N_, 256), TB, 0, stream>>>(adjm, cs);
    k_buildG<<<CDIV((long)N_ * N_, 256), TB, 0, stream>>>(adjm, rs1, cs, G1, G2);
    // stem
    k_start<<<CDIV((long)B_ * RES_C_ * N_ * RF_, 256), TB, 0, stream>>>(x, P(8), P(9), hA);
    k_skip0<<<CDIV(B_ * SKIP_C_ * N_, 256), TB, 0, stream>>>(x, P(10), P(11), skip);

    for (int i = 0; i < 3; ++i) {
        int Tin = RF_ - 6 * i, Tout = Tin - 6;
        int base = 18 + i * 24;
        int M = B_ * Tout * CONV_C_;
        IncepArgs A;
        for (int k = 0; k < 4; ++k) {
            A.fw[k] = P(base + 2 * k);     A.fb[k] = P(base + 2 * k + 1);
            A.gw[k] = P(base + 8 + 2 * k); A.gb[k] = P(base + 8 + 2 * k + 1);
        }
        long ne = (long)B_ * CONV_C_ * N_ * Tout;
        k_incep<<<CDIV(ne, 256), TB, 0, stream>>>(hA, hB, A, Tin, Tout);
        k_skip_accum<<<CDIV(B_ * SKIP_C_ * N_, 256), TB, 0, stream>>>(hB, P(base + 16), P(base + 17), skip, Tout);
        k_pack<<<CDIV(ne, 256), TB, 0, stream>>>(hB, F0, Tout);

        dim3 gg(N_ / 64, M / 64);
        // mixprop with G1
        k_gemm_prop<<<gg, TB, 0, stream>>>(G1, F0, F0, H1, M);
        k_gemm_prop<<<gg, TB, 0, stream>>>(G1, H1, F0, H2, M);
        k_mixconv<<<CDIV(ne, 256), TB, 0, stream>>>(F0, H1, H2, P(base + 18), P(base + 19), hC, nullptr, Tout, Tin, 0);
        // mixprop with G2 (adj^T), accumulate + residual
        k_gemm_prop<<<gg, TB, 0, stream>>>(G2, F0, F0, H1, M);
        k_gemm_prop<<<gg, TB, 0, stream>>>(G2, H1, F0, H2, M);
        k_mixconv<<<CDIV(ne, 256), TB, 0, stream>>>(F0, H1, H2, P(base + 20), P(base + 21), hC, hA, Tout, Tin, 1);
        // layernorm -> next layer input in hA
        k_ln_reduce<<<B_ * 256, TB, 0, stream>>>(hC, part, Tout);
        k_ln_final<<<B_, TB, 0, stream>>>(part, stats, Tout);
        k_ln_apply<<<CDIV(ne, 256), TB, 0, stream>>>(hC, stats, P(base + 22), P(base + 23), idx, hA, Tout);
    }
    // head
    k_head_skip<<<CDIV(B_ * SKIP_C_ * N_, 256), TB, 0, stream>>>(hA, P(12), P(13), skip);
    k_end1<<<CDIV(B_ * END_C_ * N_, 256), TB, 0, stream>>>(skip, P(14), P(15), e1);
    k_end2<<<CDIV(B_ * OUT_DIM_ * N_, 256), TB, 0, stream>>>(e1, P(16), P(17), (float*)d_out);
}